// IndexedDense_67542655697624
// MI455X (gfx1250) — compile-verified
//
#include <hip/hip_runtime.h>
#include <stdint.h>

// IndexedDense (16-expert dense layer) for MI455X / gfx1250.
// Grouped GEMM: block = (expert e, 64-col tile). 8 waves = 4 N-tiles x 2 M-groups.
//  - W tile (512x64 f32) staged by one TDM tensor_load_to_lds (padded rows, 65 dw).
//  - A rows gathered per 64-row chunk by 4 TDM gather-mode descriptors
//    (16-bit row indices packed from the sorted rowlist via v_readlane).
//  - Ballot-based parallel row compaction (no serial scan).
//  - Single fused K loop: 2 WMMAs (MT=2) share one B fragment per k-step;
//    the mid-row TDM pad is folded in as a scalar term ((k0>>7)&2) so the
//    whole loop keeps the fused ds_load_2addr addressing form.
//    V_WMMA_F32_16X16X4_F32, exact fp32.

#define NUM_EXPERTS 16
#define IN_DIM 512
#define UNITS 512
#define BATCH 1024

#define COLS_PER_BLOCK 64
#define THREADS 256
#define NWAVES 8
#define W_STRIDE 65          // dwords per k-row in LDS (TDM pad: +1 dw every 64 dw)
#define A_STRIDE 516         // dwords per x-row in LDS (TDM pad: +2 dw every 256 dw)
#define ROWS_PER_STAGE 64

typedef float    v2f  __attribute__((ext_vector_type(2)));
typedef float    v8f  __attribute__((ext_vector_type(8)));
typedef uint32_t u32x4 __attribute__((ext_vector_type(4)));
typedef uint32_t u32x8 __attribute__((ext_vector_type(8)));

__device__ __forceinline__ void tdm_load(u32x4 g0, u32x8 g1, u32x4 g2, u32x4 g3) {
    asm volatile("tensor_load_to_lds %0, %1, %2, %3"
                 :: "s"(g0), "s"(g1), "s"(g2), "s"(g3) : "memory");
}

// Pack lanes [l0, l0+8) of v (16-bit each) into 4 dwords.
__device__ __forceinline__ u32x4 pack_idx8(int v, int l0) {
    u32x4 r;
    #pragma unroll
    for (int i = 0; i < 4; ++i) {
        unsigned lo = (unsigned)__builtin_amdgcn_readlane(v, l0 + 2 * i) & 0xFFFFu;
        unsigned hi = (unsigned)__builtin_amdgcn_readlane(v, l0 + 2 * i + 1) & 0xFFFFu;
        r[i] = lo | (hi << 16);
    }
    return r;
}

__global__ __launch_bounds__(THREADS) void indexed_dense_kernel(
    const float* __restrict__ x, const int* __restrict__ idx,
    const float* __restrict__ ws, const float* __restrict__ bs,
    float* __restrict__ out)
{
    extern __shared__ char smem[];
    float* lds_w   = (float*)smem;                              // 512*65 dwords
    float* lds_a   = lds_w + IN_DIM * W_STRIDE;                 // 64*516 dwords
    int*   rowlist = (int*)(lds_a + ROWS_PER_STAGE * A_STRIDE); // 1024
    int*   wavecnt = rowlist + BATCH;                           // 8

    const int e    = blockIdx.x;
    const int n0   = blockIdx.y * COLS_PER_BLOCK;
    const int tid  = threadIdx.x;
    const int lane = tid & 31;
    const int wave = tid >> 5;

    // -------- TDM #1: stage W tile (512 k-rows x 64 cols) into LDS --------
    // D# (CDNA5 ISA): 2D, 4B elems, dim0_stride=512,
    // pad 1 dword per 64 dwords -> LDS k-row stride 65 dwords.
    if (wave == 0) {
        const float* gw = ws + (size_t)e * IN_DIM * UNITS + n0;
        uint64_t ga = (uint64_t)(uintptr_t)gw;
        u32x4 g0;
        g0[0] = 1u;                                          // count=1
        g0[1] = (uint32_t)(uintptr_t)lds_w;                  // lds_addr (flat addr[31:0])
        g0[2] = (uint32_t)ga;
        g0[3] = ((uint32_t)(ga >> 32) & 0x01FFFFFFu) | (2u << 30);  // type=2
        u32x8 g1;
        g1[0] = (2u << 16) | (1u << 20) | (5u << 22);        // 4B, pad_en, interval=64dw, amount=1dw
        g1[1] = ((uint32_t)IN_DIM) << 16;                    // tensor_dim0 = 512
        g1[2] = ((uint32_t)IN_DIM) << 16;                    // tensor_dim1 = 512
        g1[3] = ((uint32_t)COLS_PER_BLOCK) << 16;            // tile_dim0 = 64
        g1[4] = (uint32_t)IN_DIM;                            // tile_dim1 = 512, tile_dim2 = 0
        g1[5] = (uint32_t)UNITS;                             // tensor_dim0_stride = 512
        g1[6] = 0u; g1[7] = 0u;
        u32x4 gz = {0u, 0u, 0u, 0u};
        tdm_load(g0, g1, gz, gz);
        // No wait here: DMA overlaps the compaction below.
    }

    // -------- Ballot-based ordered row compaction (wave-parallel) --------
    // Wave w owns rows [w*128, w*128+128) in 4 ballots of 32; rowlist stays
    // sorted ascending (required by TDM gather OOB rule + determinism).
    const uint32_t lt_mask = (1u << lane) - 1u;
    uint32_t mk[4];
    int wcnt = 0;
    #pragma unroll
    for (int j = 0; j < 4; ++j) {
        int r = wave * 128 + j * 32 + lane;
        mk[j] = __builtin_amdgcn_ballot_w32(idx[r] == e);
        wcnt += __builtin_popcount(mk[j]);
    }
    if (lane == 0) wavecnt[wave] = wcnt;
    __syncthreads();
    int wbase = 0, nrows = 0;
    #pragma unroll
    for (int wv = 0; wv < NWAVES; ++wv) {
        int c = wavecnt[wv];
        if (wv < wave) wbase += c;
        nrows += c;
    }
    {
        int off = wbase;
        #pragma unroll
        for (int j = 0; j < 4; ++j) {
            if (mk[j] & (1u << lane)) {
                int rank = __builtin_popcount(mk[j] & lt_mask);
                rowlist[off + rank] = wave * 128 + j * 32 + lane;
            }
            off += __builtin_popcount(mk[j]);
        }
    }
    __syncthreads();

    // Fragment layouts (16x16x4 f32, §7.12.2): lanes 0-15 hold K={k0,k0+1},
    // lanes 16-31 K={k0+2,k0+3}; A rows / B cols = lane&15.
    const int  koff   = (lane >> 4) << 1;         // 0 | 2
    const int  nloc   = lane & 15;
    const int  ntile  = wave & 3;                 // N-tile (16 cols) within 64-col block
    const int  mgroup = wave >> 2;                // 0 | 1 -> rows [mg*32, mg*32+32)
    const int  bcol   = ntile * 16 + nloc;
    const int  u      = n0 + bcol;
    const float bias  = bs[e * UNITS + u];

    const uint32_t a_lds_base = (uint32_t)(uintptr_t)lds_a;
    const uint64_t xa = (uint64_t)(uintptr_t)x;

    // A-gather descriptor group1 (shared by all 4 gather ops):
    // gather_mode, 16-bit indices, 4B elems, pad 2 dw per 256 dw -> row stride 516 dw.
    u32x8 ag1;
    ag1[0] = (2u << 16) | (1u << 20) | (7u << 22) | (1u << 25);
    ag1[1] = ((uint32_t)IN_DIM) << 16;            // tensor_dim0 = 512
    ag1[2] = ((uint32_t)BATCH) << 16;             // tensor_dim1 = 1024 (row OOB)
    ag1[3] = ((uint32_t)IN_DIM) << 16;            // tile_dim0 = 512 (full row)
    ag1[4] = 16u;                                 // tile_dim1 = 16 valid indices
    ag1[5] = (uint32_t)IN_DIM;                    // tensor_dim0_stride = 512
    ag1[6] = 0u; ag1[7] = 0u;

    for (int base = 0; base < nrows; base += ROWS_PER_STAGE) {
        __syncthreads();                          // previous chunk's A reads done

        // -------- TDM #2..5: gather 64 x-rows (sorted, clamp-repeat pad) --------
        if (wave == 0) {
            const int last = nrows - 1;
            int j0 = base + lane;       if (j0 > last) j0 = last;
            int j1 = base + 32 + lane;  if (j1 > last) j1 = last;
            int r0 = rowlist[j0];
            int r1 = rowlist[j1];
            #pragma unroll
            for (int g = 0; g < 4; ++g) {
                int src = (g < 2) ? r0 : r1;
                int l0  = (g & 1) * 16;
                u32x4 g2 = pack_idx8(src, l0);
                u32x4 g3 = pack_idx8(src, l0 + 8);
                u32x4 g0v;
                g0v[0] = 1u | (1u << 31);                     // count=1, gather_mode, 16b idx
                g0v[1] = a_lds_base + (uint32_t)g * (16u * A_STRIDE * 4u);
                g0v[2] = (uint32_t)xa;
                g0v[3] = ((uint32_t)(xa >> 32) & 0x01FFFFFFu) | (2u << 30);
                tdm_load(g0v, ag1, g2, g3);
            }
            __builtin_amdgcn_s_wait_tensorcnt(0); // also covers the W DMA (1st iter)
        }
        __syncthreads();

        // -------- Compute: 2 M-tiles per wave sharing B fragments --------
        v8f acc[2];
        acc[0] = (v8f){0.f,0.f,0.f,0.f,0.f,0.f,0.f,0.f};
        acc[1] = (v8f){0.f,0.f,0.f,0.f,0.f,0.f,0.f,0.f};
        const int a0 = (mgroup * 32 + nloc) * A_STRIDE + koff;
        const int a1 = a0 + 16 * A_STRIDE;

        #pragma unroll 4
        for (int k0 = 0; k0 < IN_DIM; k0 += 4) {
            const int ka = k0 + ((k0 >> 7) & 2);  // scalar: skip 2-dw TDM pad at k=256
            v2f av0 = *(const v2f*)(lds_a + a0 + ka);
            v2f av1 = *(const v2f*)(lds_a + a1 + ka);
            v2f bv;
            bv.x = lds_w[(k0 + koff)     * W_STRIDE + bcol];
            bv.y = lds_w[(k0 + koff + 1) * W_STRIDE + bcol];
            acc[0] = __builtin_amdgcn_wmma_f32_16x16x4_f32(
                false, av0, false, bv, (short)0, acc[0], false, false);
            acc[1] = __builtin_amdgcn_wmma_f32_16x16x4_f32(
                false, av1, false, bv, (short)0, acc[1], false, false);
        }

        // -------- Epilogue: bias + relu, scatter rows --------
        const int mb = (lane < 16) ? 0 : 8;       // C/D: VGPR r -> M = mb + r
        #pragma unroll
        for (int t = 0; t < 2; ++t) {
            const int sbase = base + mgroup * 32 + t * 16;
            const int avail = nrows - sbase;      // m < avail (m < 16 always)
            #pragma unroll
            for (int r = 0; r < 8; ++r) {
                int m = mb + r;
                if (m < avail) {
                    int row = rowlist[sbase + m];
                    float vv = acc[t][r] + bias;
                    out[(size_t)row * UNITS + u] = fmaxf(vv, 0.f);
                }
            }
        }
    }
}

extern "C" void kernel_launch(void* const* d_in, const int* in_sizes, int n_in,
                              void* d_out, int out_size, void* d_ws, size_t ws_size,
                              hipStream_t stream) {
    (void)in_sizes; (void)n_in; (void)out_size; (void)d_ws; (void)ws_size;
    const float* x   = (const float*)d_in[0];
    const int*   idx = (const int*)d_in[1];
    const float* ws  = (const float*)d_in[2];
    const float* bs  = (const float*)d_in[3];
    float* out = (float*)d_out;

    size_t shmem = (size_t)(IN_DIM * W_STRIDE + ROWS_PER_STAGE * A_STRIDE) * sizeof(float)
                 + (size_t)(BATCH + 8) * sizeof(int);   // ~269 KB < 320 KB

    dim3 grid(NUM_EXPERTS, UNITS / COLS_PER_BLOCK);
    indexed_dense_kernel<<<grid, dim3(THREADS), shmem, stream>>>(x, idx, ws, bs, out);
}